// NodeAttentionPool_1391569404521
// MI455X (gfx1250) — compile-verified
//
#include <hip/hip_runtime.h>
#include <math.h>

#define F_DIM  128
#define RCAP   96      // rows of x cached in LDS (48 KB)
#define LCAP   2048    // logits/exp cached in LDS (8 KB)
#define NWAVES 8
#define BLOCK  256

__device__ __forceinline__ float wave_sum(float v) {
#pragma unroll
    for (int off = 16; off > 0; off >>= 1) v += __shfl_xor(v, off, 32);
    return v;
}

__global__ __launch_bounds__(BLOCK)
void node_attn_pool_kernel(const float* __restrict__ x,
                           const float* __restrict__ attn_w,
                           const float* __restrict__ attn_b,
                           const int*   __restrict__ batch_idx,
                           float* __restrict__ out,
                           int n_total)
{
    extern __shared__ float smem[];
    float* rows  = smem;                      // RCAP * F_DIM
    float* evals = rows + RCAP * F_DIM;       // LCAP
    float* acc   = evals + LCAP;              // NWAVES * F_DIM
    float* red   = acc + NWAVES * F_DIM;      // NWAVES
    int*   seg   = (int*)(red + NWAVES);      // 2

    const int g    = blockIdx.x;
    const int tid  = threadIdx.x;
    const int lane = tid & 31;
    const int wave = tid >> 5;
    const int f4   = lane << 2;               // 4 features per lane

    if (tid == 0) {   // binary-search segment bounds in sorted batch_idx
        int lo = 0, hi = n_total;
        while (lo < hi) { int mid = (lo + hi) >> 1; if (batch_idx[mid] <  g) lo = mid + 1; else hi = mid; }
        seg[0] = lo;
        hi = n_total;
        while (lo < hi) { int mid = (lo + hi) >> 1; if (batch_idx[mid] <= g) lo = mid + 1; else hi = mid; }
        seg[1] = lo;
    }
    __syncthreads();
    const int start = seg[0];
    const int len   = seg[1] - start;

    if (len <= 0) {
        if (tid < F_DIM) out[(size_t)g * F_DIM + tid] = 0.0f;
        return;
    }

    // ---- Stage up to RCAP rows of x into LDS (CDNA5 async global->LDS, ASYNCcnt)
    const int ncache = len < RCAP ? len : RCAP;
    for (int r = wave; r < ncache; r += NWAVES) {
        unsigned lds_addr = (unsigned)(size_t)(rows + r * F_DIM + f4);
        int      goff     = (start + r) * (F_DIM * 4) + (f4 * 4);
        asm volatile("global_load_async_to_lds_b128 %0, %1, %2"
                     :: "v"(lds_addr), "v"(goff), "s"(x)
                     : "memory");
    }
    asm volatile("s_wait_asynccnt 0" ::: "memory");
    __syncthreads();

    const float4 w4 = *(const float4*)(attn_w + f4);
    const float  bb = attn_b[0];

    // ---- Pass 1: logits + segment max (wave w handles rows w, w+8, ...)
    float wmax = -INFINITY;
    for (int r = wave; r < len; r += NWAVES) {
        const float* src = (r < ncache) ? (rows + r * F_DIM)
                                        : (x + (size_t)(start + r) * F_DIM);
        float4 xv = *(const float4*)(src + f4);
        float p = xv.x * w4.x + xv.y * w4.y + xv.z * w4.z + xv.w * w4.w;
        float logit = wave_sum(p) + bb;
        if (r < LCAP && lane == 0) evals[r] = logit;
        wmax = fmaxf(wmax, logit);
    }
    if (lane == 0) red[wave] = wmax;
    __syncthreads();
    float m = red[0];
#pragma unroll
    for (int i = 1; i < NWAVES; ++i) m = fmaxf(m, red[i]);
    __syncthreads();    // red[] gets reused below

    // ---- Pass 1b: exp + segment sum (thread-strided over logits)
    float psum = 0.0f;
    for (int r = tid; r < len; r += BLOCK) {
        float lv;
        if (r < LCAP) {
            lv = evals[r];
        } else {   // astronomically rare (>2048-node graph): recompute logit
            lv = bb;
            const float* xr = x + (size_t)(start + r) * F_DIM;
            for (int k = 0; k < F_DIM; ++k) lv += xr[k] * attn_w[k];
        }
        float e = expf(lv - m);
        if (r < LCAP) evals[r] = e;
        psum += e;
    }
    psum = wave_sum(psum);
    if (lane == 0) red[wave] = psum;
    __syncthreads();
    float total = 1e-16f;
#pragma unroll
    for (int i = 0; i < NWAVES; ++i) total += red[i];

    // ---- Pass 2: weighted accumulation (rows from LDS, alpha from LDS)
    float4 a = {0.0f, 0.0f, 0.0f, 0.0f};
    for (int r = wave; r < len; r += NWAVES) {
        float e;
        if (r < LCAP) {
            e = evals[r];
        } else {
            const float* xr = x + (size_t)(start + r) * F_DIM;
            float4 xv = *(const float4*)(xr + f4);
            float p = xv.x * w4.x + xv.y * w4.y + xv.z * w4.z + xv.w * w4.w;
            e = expf(wave_sum(p) + bb - m);
        }
        float alpha = e / total;
        const float* src = (r < ncache) ? (rows + r * F_DIM)
                                        : (x + (size_t)(start + r) * F_DIM);
        float4 xv = *(const float4*)(src + f4);
        a.x += alpha * xv.x; a.y += alpha * xv.y;
        a.z += alpha * xv.z; a.w += alpha * xv.w;
    }
    *(float4*)(acc + wave * F_DIM + f4) = a;
    __syncthreads();

    if (tid < F_DIM) {
        float s = 0.0f;
#pragma unroll
        for (int w2 = 0; w2 < NWAVES; ++w2) s += acc[w2 * F_DIM + tid];
        out[(size_t)g * F_DIM + tid] = s;
    }
}

extern "C" void kernel_launch(void* const* d_in, const int* in_sizes, int n_in,
                              void* d_out, int out_size, void* d_ws, size_t ws_size,
                              hipStream_t stream) {
    const float* x      = (const float*)d_in[0];
    const float* attn_w = (const float*)d_in[1];
    const float* attn_b = (const float*)d_in[2];
    const int*   bidx   = (const int*)d_in[3];
    // d_in[4] = num_graphs (device scalar); derive counts host-side instead:
    const int n_total    = in_sizes[3];
    const int num_graphs = out_size / F_DIM;

    const size_t lds_bytes =
        (size_t)(RCAP * F_DIM + LCAP + NWAVES * F_DIM + NWAVES) * sizeof(float)
        + 2 * sizeof(int);

    node_attn_pool_kernel<<<num_graphs, BLOCK, lds_bytes, stream>>>(
        x, attn_w, attn_b, bidx, (float*)d_out, n_total);
}